// ChunkLayer_292057776375
// MI455X (gfx1250) — compile-verified
//
#include <hip/hip_runtime.h>
#include <stdint.h>

// Problem shape fixed by the reference: B=8, L=8192, D=1024 (fp32).
#define B_ 8
#define L_ 8192
#define D_ 1024
#define GATHER_THREADS 256          // 8 waves (wave32); 256*16B == one 4096B row

// ---------------------------------------------------------------------------
// Kernel 1: per-batch stable partition index build.
// token_idx = i + (!boundary)*L ; stable argsort == boundary tokens first (in
// order), then the rest (in order). Equivalent to prefix sums of the mask.
// ---------------------------------------------------------------------------
__global__ __launch_bounds__(256)
void build_indices(const int* __restrict__ boundary,
                   int* __restrict__ idxbuf,     // [B, L]
                   int* __restrict__ numtok) {   // [B]
    const int b    = blockIdx.x;
    const int t    = threadIdx.x;
    const int EPT  = L_ / 256;                   // 32 contiguous elems/thread
    const int base = b * L_;

    __shared__ int scan[256];

    int cnt = 0;
    #pragma unroll 8
    for (int i = 0; i < EPT; ++i)
        cnt += (boundary[base + t * EPT + i] != 0);
    scan[t] = cnt;
    __syncthreads();

    // Inclusive Hillis-Steele scan over 256 per-thread counts.
    for (int off = 1; off < 256; off <<= 1) {
        int add = (t >= off) ? scan[t - off] : 0;
        int v   = scan[t];
        __syncthreads();
        scan[t] = v + add;
        __syncthreads();
    }

    const int total = scan[255];                 // num_tokens[b]
    int bprefix = (t == 0) ? 0 : scan[t - 1];    // exclusive boundary count

    int gi = t * EPT;
    for (int i = 0; i < EPT; ++i, ++gi) {
        const int isb = (boundary[base + gi] != 0);
        // boundary token -> slot bprefix; non-boundary -> total + (#nonb before)
        const int pos = isb ? bprefix : (total + (gi - bprefix));
        idxbuf[base + pos] = gi;
        bprefix += isb;
    }
    if (t == 0) numtok[b] = total;
}

// ---------------------------------------------------------------------------
// Kernel 2: bandwidth kernel. Gather S rows per batch via the CDNA5 async
// global<->LDS DMA path (ASYNCcnt), double-buffered in LDS.
//
// Per lane per row: one global_load_async_to_lds_b128 (16B) and one
// global_store_async_from_lds_b128 (16B); 256 lanes cover the 4096B row.
// Each lane stores exactly the LDS bytes its own wave's load wrote, so a
// per-wave s_wait_asynccnt 0 is the only synchronization required. The wait
// in iteration i-1 also drains the store from iteration i-2 that used the same
// LDS buffer, making the double-buffer reuse safe while overlapping the HBM
// read of row i with the HBM write of row i-1.
//
// Footprint (256 MiB read + ~140 MiB write, zero reuse) exceeds the 192 MB
// L2, so both sides use non-temporal hints to keep the stream from rinsing
// the cache.
// ---------------------------------------------------------------------------
__global__ __launch_bounds__(GATHER_THREADS)
void gather_rows(const float* __restrict__ hs,     // [B, L, D]
                 const int*  __restrict__ idxbuf,  // [B, L]
                 const int*  __restrict__ numtok,  // [B]
                 float* __restrict__ out_h,        // [B, S, D]
                 float* __restrict__ out_m,        // [B, S]
                 int S, int nrows) {
    __shared__ __align__(16) unsigned char smem[2][GATHER_THREADS * 16]; // 8KB

    const int t = threadIdx.x;
    // Truncating a flat pointer to LDS gives the 32-bit LDS byte address
    // (flat LDS addr = {SHARED_BASE, lds_offset}).
    const unsigned lds0 = (unsigned)(uintptr_t)(&smem[0][t * 16]);
    const unsigned lds1 = (unsigned)(uintptr_t)(&smem[1][t * 16]);

    int it = 0;
    for (int row = blockIdx.x; row < nrows; row += gridDim.x, ++it) {
        const int b   = row / S;
        const int j   = row - b * S;
        const int src = idxbuf[b * L_ + j];

        const uint64_t gsrc = (uint64_t)(uintptr_t)(hs + ((size_t)b * L_ + src) * D_)
                              + (unsigned)(t * 16);
        const uint64_t gdst = (uint64_t)(uintptr_t)(out_h + (size_t)row * D_)
                              + (unsigned)(t * 16);
        const unsigned lbuf = (it & 1) ? lds1 : lds0;

        asm volatile("global_load_async_to_lds_b128 %0, %1, off th:TH_LOAD_NT"
                     :: "v"(lbuf), "v"(gsrc) : "memory");
        asm volatile("s_wait_asynccnt 0x0" ::: "memory");
        asm volatile("global_store_async_from_lds_b128 %0, %1, off th:TH_STORE_NT"
                     :: "v"(gdst), "v"(lbuf) : "memory");

        if (t == 0)
            __builtin_nontemporal_store((j < numtok[b]) ? 1.0f : 0.0f,
                                        out_m + row);
    }
    asm volatile("s_wait_asynccnt 0x0" ::: "memory");
}

// ---------------------------------------------------------------------------
extern "C" void kernel_launch(void* const* d_in, const int* in_sizes, int n_in,
                              void* d_out, int out_size, void* d_ws, size_t ws_size,
                              hipStream_t stream) {
    (void)in_sizes; (void)n_in; (void)ws_size;

    const float* hs       = (const float*)d_in[0];
    const int*   boundary = (const int*)d_in[1];
    // d_in[2] (mask) is all-ones and unused by the reference math.

    // out_size = B*S*D + B*S  ->  S
    const int S = out_size / (B_ * (D_ + 1));

    int*   idxbuf = (int*)d_ws;                  // B*L ints = 256 KiB
    int*   numtok = idxbuf + (size_t)B_ * L_;    // B ints
    float* out_h  = (float*)d_out;               // [B, S, D]
    float* out_m  = out_h + (size_t)B_ * S * D_; // [B, S] as float 0/1

    build_indices<<<B_, 256, 0, stream>>>(boundary, idxbuf, numtok);

    const int nrows = B_ * S;
    int grid = nrows < 8192 ? nrows : 8192;
    if (grid < 1) grid = 1;
    gather_rows<<<grid, GATHER_THREADS, 0, stream>>>(hs, idxbuf, numtok,
                                                     out_h, out_m, S, nrows);
}